// StructureRefiner_32229434589573
// MI455X (gfx1250) — compile-verified
//
#include <hip/hip_runtime.h>
#include <math.h>

// Problem constants (from reference)
#define BB 4
#define SS 1024
#define DD 768
#define HH 384
#define NROWS (BB * SS)          // 4096
#define NX (NROWS * 3)           // 12288 floats in x / m / v / g / delta
#define BONDL 3.8f
#define CONTACT 8.0f

typedef __attribute__((ext_vector_type(2))) float v2f;
typedef __attribute__((ext_vector_type(8))) float v8f;

// ---------------------------------------------------------------------------
// Kernel 1: backbone MLP via fp32 WMMA (16x16x4), fused 2nd layer + delta.
// One block = 16 rows of X. Dynamic LDS: Xs[16][772] + Hs[16][388] + Ang[16][3]
// ---------------------------------------------------------------------------
#define XPITCH 772   // 768 + 4 pad (banks)
#define HPITCH 388   // 384 + 4 pad

__global__ __launch_bounds__(256) void mlp_delta_kernel(
    const float* __restrict__ X,    // [4096, 768]
    const float* __restrict__ W1,   // [768, 384]
    const float* __restrict__ b1,   // [384]
    const float* __restrict__ W2,   // [384, 3]
    const float* __restrict__ b2,   // [3]
    float* __restrict__ delta)      // [4096, 3]
{
    extern __shared__ float smem[];
    float* Xs  = smem;                       // 16*772
    float* Hs  = smem + 16 * XPITCH;         // 16*388
    float* Ang = Hs + 16 * HPITCH;           // 16*3

    const int m0  = blockIdx.x * 16;
    const int tid = threadIdx.x;

    // Stage X tile (rows m0..m0+15 are one contiguous 12288-float chunk).
    for (int idx = tid; idx < 16 * DD; idx += 256) {
        int r = idx / DD, c = idx - r * DD;
        Xs[r * XPITCH + c] = X[(size_t)m0 * DD + idx];
    }
    __syncthreads();

    const int wave = tid >> 5;
    const int lane = tid & 31;
    const int lrow = lane & 15;     // A row / B-C-D column within tile
    const int kgrp = lane >> 4;     // K group (0: k+0,k+1 ; 1: k+2,k+3)

    // Each of 8 waves computes 3 of the 24 column tiles of H = relu(X*W1+b1).
    for (int nt = wave; nt < HH / 16; nt += 8) {
        const int n0 = nt * 16;
        v8f acc = {};
        for (int k = 0; k < DD; k += 4) {
            const int ka = k + kgrp * 2;
            v2f a, bf;
            // A 16x4 f32 layout: lanes 0-15 hold K=0,1; lanes 16-31 hold K=2,3
            a.x = Xs[lrow * XPITCH + ka + 0];
            a.y = Xs[lrow * XPITCH + ka + 1];
            // B 4x16 f32: rows striped across lanes, same K grouping as A
            bf.x = W1[(size_t)(ka + 0) * HH + n0 + lrow];
            bf.y = W1[(size_t)(ka + 1) * HH + n0 + lrow];
            acc = __builtin_amdgcn_wmma_f32_16x16x4_f32(
                false, a, false, bf, (short)0, acc, false, false);
        }
        // C/D layout: vgpr r -> row (kgrp*8 + r), column lrow. Bias + ReLU.
        const int  ncol = n0 + lrow;
        const float bia = b1[ncol];
        #pragma unroll
        for (int r = 0; r < 8; ++r) {
            float hv = acc[r] + bia;
            Hs[(kgrp * 8 + r) * HPITCH + ncol] = hv > 0.0f ? hv : 0.0f;
        }
    }
    __syncthreads();

    // Second layer: 16 rows x 3 outputs, one thread each (384-long dots in LDS)
    if (tid < 48) {
        const int row = tid / 3, c = tid - row * 3;
        float s = b2[c];
        for (int n = 0; n < HH; ++n) s += Hs[row * HPITCH + n] * W2[n * 3 + c];
        Ang[row * 3 + c] = s;
    }
    __syncthreads();

    // delta = [cos(phi)cos(psi), sin(phi)cos(psi), sin(psi)] * BOND; row 0 of
    // each chain forced to zero.
    if (tid < 16) {
        const int gr = m0 + tid;
        const int sIdx = gr & (SS - 1);
        float dx = 0.0f, dy = 0.0f, dz = 0.0f;
        if (sIdx != 0) {
            float sph, cph, sps, cps;
            __sincosf(Ang[tid * 3 + 0], &sph, &cph);
            __sincosf(Ang[tid * 3 + 1], &sps, &cps);
            dx = cph * cps * BONDL;
            dy = sph * cps * BONDL;
            dz = sps * BONDL;
        }
        delta[(size_t)gr * 3 + 0] = dx;
        delta[(size_t)gr * 3 + 1] = dy;
        delta[(size_t)gr * 3 + 2] = dz;
    }
}

// ---------------------------------------------------------------------------
// Kernel 2: per-(batch,dim) cumulative sum along S via wave32 scan.
// grid = 12 blocks of 32 threads.
// ---------------------------------------------------------------------------
__global__ void cumsum_kernel(const float* __restrict__ delta,
                              float* __restrict__ x)
{
    const int b   = blockIdx.x / 3;
    const int dim = blockIdx.x - b * 3;
    const int lane = threadIdx.x;
    const float* db = delta + (size_t)b * SS * 3 + dim;
    float*       xb = x     + (size_t)b * SS * 3 + dim;

    const int base = lane * 32;
    float chunk = 0.0f;
    for (int t = 0; t < 32; ++t) chunk += db[(size_t)(base + t) * 3];

    float incl = chunk;
    for (int off = 1; off < 32; off <<= 1) {
        float n = __shfl_up(incl, off, 32);
        if (lane >= off) incl += n;
    }
    float run = incl - chunk;   // exclusive prefix of this lane's chunk
    for (int t = 0; t < 32; ++t) {
        run += db[(size_t)(base + t) * 3];
        xb[(size_t)(base + t) * 3] = run;
    }
}

// ---------------------------------------------------------------------------
// Kernel 3: symmetrize contact map, cs[b,i,j] = c[b,i,j] + c[b,j,i]
// (tiled LDS transpose so both global reads are coalesced). Done once;
// the 50-step loop then does a single coalesced L2 read per pair.
// ---------------------------------------------------------------------------
__global__ __launch_bounds__(1024) void sym_kernel(const float* __restrict__ c,
                                                   float* __restrict__ cs)
{
    __shared__ float tile[32][33];
    const int b  = blockIdx.z;
    const int i0 = blockIdx.y * 32, j0 = blockIdx.x * 32;
    const int tx = threadIdx.x, ty = threadIdx.y;
    const float* cb = c + (size_t)b * SS * SS;

    tile[ty][tx] = cb[(size_t)(j0 + ty) * SS + (i0 + tx)];
    __syncthreads();
    const float a = cb[(size_t)(i0 + ty) * SS + (j0 + tx)];
    cs[(size_t)b * SS * SS + (size_t)(i0 + ty) * SS + (j0 + tx)] = a + tile[tx][ty];
}

// ---------------------------------------------------------------------------
// Kernel 4: gradient of contact loss. One wave per row i; block stages the
// whole batch's x (12 KB) in LDS. grid = 512 blocks of 256.
// ---------------------------------------------------------------------------
__global__ __launch_bounds__(256) void grad_kernel(
    const float* __restrict__ x,
    const float* __restrict__ cmap,   // symmetrized (sym=1) or raw (sym=0)
    float* __restrict__ g,
    int sym)
{
    __shared__ float xs[SS * 3];
    const int b     = blockIdx.x >> 7;           // 128 blocks per batch
    const int rbase = (blockIdx.x & 127) * 8;
    const int tid   = threadIdx.x;

    for (int idx = tid; idx < SS * 3; idx += 256)
        xs[idx] = x[(size_t)b * SS * 3 + idx];
    __syncthreads();

    const int wave = tid >> 5, lane = tid & 31;
    const int i = rbase + wave;
    const float xi0 = xs[i * 3 + 0], xi1 = xs[i * 3 + 1], xi2 = xs[i * 3 + 2];
    const float* crow = cmap + (size_t)b * SS * SS + (size_t)i * SS;
    const float* ccol = cmap + (size_t)b * SS * SS + i;

    float a0 = 0.0f, a1 = 0.0f, a2 = 0.0f;
    for (int j = lane; j < SS; j += 32) {
        float w = crow[j];
        if (!sym) w += ccol[(size_t)j * SS];
        const float dx = xi0 - xs[j * 3 + 0];
        const float dy = xi1 - xs[j * 3 + 1];
        const float dz = xi2 - xs[j * 3 + 2];
        const float s  = dx * dx + dy * dy + dz * dz + 1e-12f;
        const float ri = __frsqrt_rn(s);      // 1/d
        const float d  = s * ri;              // d
        const float t  = (d >= CONTACT ? w : -w) * ri;
        a0 += t * dx; a1 += t * dy; a2 += t * dz;
    }
    for (int off = 16; off > 0; off >>= 1) {
        a0 += __shfl_down(a0, off, 32);
        a1 += __shfl_down(a1, off, 32);
        a2 += __shfl_down(a2, off, 32);
    }
    if (lane == 0) {
        const float scale = 1.0f / (float)((size_t)BB * SS * SS);
        const size_t o = ((size_t)b * SS + i) * 3;
        g[o + 0] = a0 * scale;
        g[o + 1] = a1 * scale;
        g[o + 2] = a2 * scale;
    }
}

// ---------------------------------------------------------------------------
// Kernel 5: Adam update (bias-correction factors precomputed on host)
// ---------------------------------------------------------------------------
__global__ void adam_kernel(float* __restrict__ x, const float* __restrict__ g,
                            float* __restrict__ m, float* __restrict__ v,
                            float c1, float c2)
{
    const int i = blockIdx.x * blockDim.x + threadIdx.x;
    if (i >= NX) return;
    const float gi = g[i];
    const float mi = 0.9f   * m[i] + 0.1f   * gi;
    const float vi = 0.999f * v[i] + 0.001f * gi * gi;
    m[i] = mi; v[i] = vi;
    x[i] -= 0.01f * (mi * c1) / (sqrtf(vi * c2) + 1e-8f);
}

__global__ void zero_kernel(float* __restrict__ p, int n)
{
    const int i = blockIdx.x * blockDim.x + threadIdx.x;
    if (i < n) p[i] = 0.0f;
}

// ---------------------------------------------------------------------------
extern "C" void kernel_launch(void* const* d_in, const int* in_sizes, int n_in,
                              void* d_out, int out_size, void* d_ws, size_t ws_size,
                              hipStream_t stream)
{
    (void)in_sizes; (void)n_in; (void)out_size;
    const float* Xb   = (const float*)d_in[0];  // backbone_features
    // d_in[1] side_chain_features: dead in reference
    const float* cmap = (const float*)d_in[2];  // contact_map
    const float* Wb1  = (const float*)d_in[3];
    const float* bb1  = (const float*)d_in[4];
    const float* Wb2  = (const float*)d_in[5];
    const float* bb2  = (const float*)d_in[6];
    // d_in[7..10] side-chain weights: dead

    float* x  = (float*)d_out;                  // live optimizer state = output
    char*  ws = (char*)d_ws;

    const size_t CS_BYTES  = (size_t)BB * SS * SS * sizeof(float);  // 16 MB
    const size_t VEC_BYTES = (size_t)NX * sizeof(float);            // 48 KB
    const bool use_sym = ws_size >= CS_BYTES + 4 * VEC_BYTES;

    float *cs, *m, *v, *g, *delta;
    if (use_sym) {
        cs    = (float*)ws;
        m     = (float*)(ws + CS_BYTES);
    } else {
        cs    = nullptr;
        m     = (float*)ws;
    }
    v     = m + NX;
    g     = v + NX;
    delta = g + NX;

    // 1) MLP -> delta  (fp32 WMMA; 74,432 B dynamic LDS, within 320 KB/WGP)
    const size_t mlp_lds = (size_t)(16 * XPITCH + 16 * HPITCH + 48) * sizeof(float);
    mlp_delta_kernel<<<NROWS / 16, 256, mlp_lds, stream>>>(Xb, Wb1, bb1, Wb2, bb2, delta);

    // 2) cumsum -> x0 (writes every element of d_out)
    cumsum_kernel<<<BB * 3, 32, 0, stream>>>(delta, x);

    // 3) zero Adam state (m,v contiguous)
    zero_kernel<<<(2 * NX + 255) / 256, 256, 0, stream>>>(m, 2 * NX);

    // 4) symmetrize contact map once (stays hot in 192 MB L2)
    const float* cm_used = cmap;
    int sym = 0;
    if (use_sym) {
        sym_kernel<<<dim3(SS / 32, SS / 32, BB), dim3(32, 32), 0, stream>>>(cmap, cs);
        cm_used = cs; sym = 1;
    }

    // 5) 50 Adam steps; device-wide dependency => grad + update per step
    for (int step = 1; step <= 50; ++step) {
        grad_kernel<<<BB * (SS / 8), 256, 0, stream>>>(x, cm_used, g, sym);
        const float c1 = 1.0f / (1.0f - powf(0.9f,   (float)step));
        const float c2 = 1.0f / (1.0f - powf(0.999f, (float)step));
        adam_kernel<<<(NX + 255) / 256, 256, 0, stream>>>(x, g, m, v, c1, c2);
    }
}